// GaussianAttention_16956530884844
// MI455X (gfx1250) — compile-verified
//
#include <hip/hip_runtime.h>
#include <hip/hip_bf16.h>

// GaussianAttention: y[b,s,i,j] = exp(-dist(b,i,j)^2 / (2*sigma_s^2))
// dist = L1 distance over D=3 dims; sigma_s = multiplier * base^s.
// Store-bandwidth-bound: 256 MB fp32 out -> ~11us floor @ 23.3 TB/s.
// CDNA5 paths used: async global->LDS DMA (ASYNCcnt) for input staging,
// NT b128 stores for the 256MB output stream (bigger than the 192MB L2).

#define N_PTS        2048
#define N_BATCH      2
#define N_HEADS      8
#define DIMS         3
#define ROWS_PER_BLK 16
#define TPB          256

typedef float v4f __attribute__((ext_vector_type(4)));

__device__ __forceinline__ float fast_exp2(float v) {
#if __has_builtin(__builtin_amdgcn_exp2f)
  return __builtin_amdgcn_exp2f(v);   // raw v_exp_f32 (2^x)
#else
  return exp2f(v);
#endif
}

__global__ __launch_bounds__(TPB, 1) void gaussian_attn_kernel(
    const float* __restrict__ x,      // [B, N, D]
    const float* __restrict__ mult,   // [1]
    const float* __restrict__ base,   // [1]
    float* __restrict__ out)          // [B, H, N, N]
{
  __shared__ __align__(16) float xs[N_PTS * DIMS];  // 24 KB AoS mirror of x[b]

  const int tid            = threadIdx.x;
  const int blocksPerBatch = N_PTS / ROWS_PER_BLK;  // 128
  const int b              = blockIdx.x / blocksPerBatch;
  const int i0             = (blockIdx.x % blocksPerBatch) * ROWS_PER_BLK;

  // ---- Stage x[b,:,:] (24 KB contiguous) into LDS via async global->LDS DMA.
  // Generic LDS pointers on AMDGPU carry the wave-relative LDS byte offset in
  // their low 32 bits (ISA 10.2), which is exactly what
  // the VDST VGPR of global_load_async_to_lds expects.
  {
    const float* src   = x + (size_t)b * (N_PTS * DIMS);
    const int nChunks  = (N_PTS * DIMS) / 4;        // 1536 x 16-byte chunks
    for (int t = tid; t < nChunks; t += TPB) {
      unsigned lds_off =
          (unsigned)(unsigned long long)(const void*)&xs[t * 4];
      unsigned long long gaddr =
          (unsigned long long)(const void*)(src + t * 4);
      asm volatile("global_load_async_to_lds_b128 %0, %1, off"
                   :: "v"(lds_off), "v"(gaddr)
                   : "memory");
    }
    asm volatile("s_wait_asynccnt 0" ::: "memory");  // drain this wave's ASYNCcnt
  }
  __syncthreads();                                   // all waves' DMAs visible

  // ---- Per-head exponent constants: exp(-d2/(2s^2)) == exp2(c_s * d2),
  //      c_s = -log2(e)/(2*sigma_s^2). Uniform across the wave; 8 rcps total.
  float c[N_HEADS];
  {
    const float m  = mult[0];
    const float bs = base[0];
    float sig = m;
#pragma unroll
    for (int s = 0; s < N_HEADS; ++s) {
      c[s] = -1.44269504088896340736f / (2.0f * sig * sig);
      sig *= bs;
    }
  }

  const v4f*   xs4 = (const v4f*)xs;
  const size_t NN2 = (size_t)N_PTS * N_PTS;

  for (int r = 0; r < ROWS_PER_BLK; ++r) {
    const int   i   = i0 + r;
    const float xi0 = xs[i * 3 + 0];   // same address all lanes -> LDS broadcast
    const float xi1 = xs[i * 3 + 1];
    const float xi2 = xs[i * 3 + 2];
    float* orow = out + ((size_t)b * N_HEADS * N_PTS + (size_t)i) * N_PTS;

    // 4 consecutive j per thread: 12 contiguous LDS floats = 3 x ds_load_b128.
    for (int jb = tid * 4; jb < N_PTS; jb += TPB * 4) {
      const int q  = (jb * 3) >> 2;
      const v4f A  = xs4[q + 0];
      const v4f Bv = xs4[q + 1];
      const v4f Cv = xs4[q + 2];

      float d0 = fabsf(A.x  - xi0) + fabsf(A.y  - xi1) + fabsf(A.z  - xi2);
      float d1 = fabsf(A.w  - xi0) + fabsf(Bv.x - xi1) + fabsf(Bv.y - xi2);
      float d2 = fabsf(Bv.z - xi0) + fabsf(Bv.w - xi1) + fabsf(Cv.x - xi2);
      float d3 = fabsf(Cv.y - xi0) + fabsf(Cv.z - xi1) + fabsf(Cv.w - xi2);

      const v4f dd = { d0 * d0, d1 * d1, d2 * d2, d3 * d3 };

#pragma unroll
      for (int s = 0; s < N_HEADS; ++s) {
        v4f o;
        o.x = fast_exp2(c[s] * dd.x);
        o.y = fast_exp2(c[s] * dd.y);
        o.z = fast_exp2(c[s] * dd.z);
        o.w = fast_exp2(c[s] * dd.w);
        // 256MB streaming output > 192MB L2: non-temporal b128 store.
        __builtin_nontemporal_store(o, (v4f*)(orow + (size_t)s * NN2 + jb));
      }
    }
  }
}

extern "C" void kernel_launch(void* const* d_in, const int* in_sizes, int n_in,
                              void* d_out, int out_size, void* d_ws, size_t ws_size,
                              hipStream_t stream) {
  const float* x    = (const float*)d_in[0];
  const float* mlt  = (const float*)d_in[1];
  const float* base = (const float*)d_in[2];
  float*       out  = (float*)d_out;

  dim3 grid(N_BATCH * (N_PTS / ROWS_PER_BLK));  // 256 blocks
  dim3 block(TPB);                              // 8 wave32s
  gaussian_attn_kernel<<<grid, block, 0, stream>>>(x, mlt, base, out);

  (void)in_sizes; (void)n_in; (void)out_size; (void)d_ws; (void)ws_size;
}